// LWMMamba_20023137534115
// MI455X (gfx1250) — compile-verified
//
#include <hip/hip_runtime.h>
#include <math.h>

#define D_MODEL 256
#define SEQ_LEN 2048
#define BATCH 8
#define MROWS (BATCH * SEQ_LEN)   // 16384 rows
#define D_STATE 16
#define LN_EPS 1e-5f

typedef __attribute__((ext_vector_type(2))) float v2f;
typedef __attribute__((ext_vector_type(8))) float v8f;

enum { ACT_NONE = 0, ACT_GELU = 1, ACT_SILU = 2 };

// ---------------------------------------------------------------------------
// CDNA5 async copy: global -> LDS, 16B per lane, tracked by ASYNCcnt.
// saddr form: mem_addr = SGPR64 base + VGPR32 byte offset (ISA 15.18.3).
// LDS dest address = low 32 bits of the flat shared pointer.
// ---------------------------------------------------------------------------
__device__ __forceinline__ void async_ld16(const void* lds_dst, unsigned gbyte_off,
                                           const float* gbase)
{
    unsigned lds_off = (unsigned)(unsigned long long)lds_dst;
    asm volatile("global_load_async_to_lds_b128 %0, %1, %2 offset:0"
                 :: "v"(lds_off), "v"(gbyte_off), "s"(gbase)
                 : "memory");
}

// ---------------------------------------------------------------------------
// GEMM: C = act(A[M,K] @ W[K,N] + bias[N]) using V_WMMA_F32_16X16X4_F32.
// Block tile 128x64, 8 waves; wave w -> rows [w*16, w*16+16), all 64 cols
// (4 WMMA n-tiles). K consumed in chunks of 16 (4 wmma k-steps).
// Double-buffered LDS fed by GLOBAL_LOAD_ASYNC_TO_LDS_B128: chunk k+1 streams
// in (3 async ops/thread) while chunk k feeds 16 WMMAs per wave.
// A-frag (16x4 f32): lanes 0-15 = M rows with K pair {0,1}; lanes 16-31 K {2,3}.
// B-frag (4x16 f32): lanes 0-15 = N cols with K rows {0,1}; lanes 16-31 K {2,3}.
// D (16x16 f32): vgpr r = row r (lanes 0-15) / row r+8 (lanes 16-31), N = lane&15.
// ---------------------------------------------------------------------------
template <int ACT>
__global__ __launch_bounds__(256)
void gemm_bias_act(const float* __restrict__ A, const float* __restrict__ W,
                   const float* __restrict__ bias, float* __restrict__ C,
                   int N, int K)
{
    // Row strides: As 20 dwords (80B, 16B-aligned, conflict-free reads),
    // Bs 64 dwords (256B). Buffer strides are multiples of 16B.
    __shared__ alignas(16) float As[2][128][20];
    __shared__ alignas(16) float Bs[2][16][64];

    const int tid  = threadIdx.x;
    const int wave = tid >> 5;
    const int lane = tid & 31;
    const int l15  = lane & 15;
    const int kh   = lane >> 4;          // half-wave selects K pair
    const int m0 = blockIdx.y * 128;
    const int n0 = blockIdx.x * 64;

    v8f acc[4];
    #pragma unroll
    for (int i = 0; i < 4; ++i) acc[i] = (v8f){0.f,0.f,0.f,0.f,0.f,0.f,0.f,0.f};

    const int nch = K >> 4;

    // Issue one K-chunk (3 async b128 per thread: 2 x A-tile, 1 x B-tile).
    auto issue_chunk = [&](int ch, int buf) {
        const int k0 = ch << 4;
        #pragma unroll
        for (int i = 0; i < 2; ++i) {
            int idx = tid + (i << 8);
            int r   = idx >> 2;          // 0..127
            int c4  = (idx & 3) << 2;    // 0,4,8,12
            async_ld16(&As[buf][r][c4],
                       (unsigned)(((m0 + r) * K + (k0 + c4)) * 4), A);
        }
        {
            int r  = tid >> 4;           // 0..15
            int c4 = (tid & 15) << 2;    // 0..60
            async_ld16(&Bs[buf][r][c4],
                       (unsigned)(((k0 + r) * N + (n0 + c4)) * 4), W);
        }
    };

    issue_chunk(0, 0);

    for (int ch = 0; ch < nch; ++ch) {
        const int buf = ch & 1;
        if (ch + 1 < nch) {
            issue_chunk(ch + 1, buf ^ 1);
            // chunk `ch` = oldest 3 ops; retire them, keep prefetch in flight
            asm volatile("s_wait_asynccnt 0x3" ::: "memory");
        } else {
            asm volatile("s_wait_asynccnt 0x0" ::: "memory");
        }
        __syncthreads();   // publish LDS writes across all waves

        const int arow = wave * 16 + l15;
        #pragma unroll
        for (int kk = 0; kk < 16; kk += 4) {
            v2f a = *(const v2f*)&As[buf][arow][kk + kh * 2];   // contiguous K pair
            #pragma unroll
            for (int nt = 0; nt < 4; ++nt) {
                v2f b;
                b.x = Bs[buf][kk + kh * 2 + 0][nt * 16 + l15];
                b.y = Bs[buf][kk + kh * 2 + 1][nt * 16 + l15];
                acc[nt] = __builtin_amdgcn_wmma_f32_16x16x4_f32(
                    false, a, false, b, (short)0, acc[nt], false, false);
            }
        }
        __syncthreads();   // all waves done reading before buffer is rewritten
    }

    #pragma unroll
    for (int nt = 0; nt < 4; ++nt) {
        const int col = n0 + nt * 16 + l15;
        const float bv = bias[col];
        #pragma unroll
        for (int r = 0; r < 8; ++r) {
            const int row = m0 + wave * 16 + kh * 8 + r;
            float x = acc[nt][r] + bv;
            if (ACT == ACT_GELU)      x = 0.5f * x * (1.0f + erff(x * 0.70710678118654752f));
            else if (ACT == ACT_SILU) x = x / (1.0f + expf(-x));
            C[(size_t)row * N + col] = x;
        }
    }
}

template <int ACT>
static inline void launch_gemm(const float* A, const float* W, const float* b,
                               float* C, int N, int K, hipStream_t s)
{
    dim3 grid(N / 64, MROWS / 128);
    gemm_bias_act<ACT><<<grid, dim3(256), 0, s>>>(A, W, b, C, N, K);
}

// ---------------------------------------------------------------------------
// Fused depthwise conv (4-tap causal) -> diagonal-A SSM recurrence (x2) ->
// residual -> SiLU-gate multiply -> accumulate into combined.
// One thread per (batch, channel); lane = channel => coalesced V/G/OUT access.
// Exact equivalent of the reference FFT convolution (h = a*h + dB*x; y = C.h + x).
// ---------------------------------------------------------------------------
__global__ __launch_bounds__(256)
void conv_ssm_gate(const float* __restrict__ V, const float* __restrict__ G,
                   const float* __restrict__ cw1, const float* __restrict__ cb1,
                   const float* __restrict__ A1,  const float* __restrict__ dl1,
                   const float* __restrict__ B1m, const float* __restrict__ C1m,
                   const float* __restrict__ cw2, const float* __restrict__ cb2,
                   const float* __restrict__ A2,  const float* __restrict__ dl2,
                   const float* __restrict__ B2m, const float* __restrict__ C2m,
                   float* __restrict__ OUT, int add)
{
    const int b = blockIdx.x;
    const int d = threadIdx.x;

    float w1[4], w2[4];
    #pragma unroll
    for (int k = 0; k < 4; ++k) { w1[k] = cw1[d * 4 + k]; w2[k] = cw2[d * 4 + k]; }
    const float bb1 = cb1[d], bb2 = cb2[d];

    float x1 = dl1[d];
    const float dt1 = fminf((x1 > 20.f) ? x1 : log1pf(expf(x1)), 10.f);
    float x2 = dl2[d];
    const float dt2 = fminf((x2 > 20.f) ? x2 : log1pf(expf(x2)), 10.f);

    float a1[D_STATE], a2[D_STATE], Bb1[D_STATE], Bb2[D_STATE];
    float Cc1[D_STATE], Cc2[D_STATE], h1[D_STATE], h2[D_STATE];
    #pragma unroll
    for (int n = 0; n < D_STATE; ++n) {
        a1[n]  = A1[n * D_STATE + n];
        a2[n]  = A2[n * D_STATE + n];
        Bb1[n] = dt1 * B1m[d * D_STATE + n];
        Bb2[n] = dt2 * B2m[d * D_STATE + n];
        Cc1[n] = C1m[n * D_MODEL + d];
        Cc2[n] = C2m[n * D_MODEL + d];
        h1[n] = 0.f; h2[n] = 0.f;
    }

    const size_t base = (size_t)b * SEQ_LEN * D_MODEL + d;
    float vm1 = 0.f, vm2 = 0.f, vm3 = 0.f;   // v[t-1], v[t-2], v[t-3]

    for (int t = 0; t < SEQ_LEN; ++t) {
        const size_t idx = base + (size_t)t * D_MODEL;
        const float vt = V[idx];
        const float c1 = w1[0]*vm3 + w1[1]*vm2 + w1[2]*vm1 + w1[3]*vt + bb1;
        const float c2 = w2[0]*vm3 + w2[1]*vm2 + w2[2]*vm1 + w2[3]*vt + bb2;
        vm3 = vm2; vm2 = vm1; vm1 = vt;

        float y1 = 0.f, y2 = 0.f;
        #pragma unroll
        for (int n = 0; n < D_STATE; ++n) {
            h1[n] = fmaf(a1[n], h1[n], Bb1[n] * c1);
            y1    = fmaf(Cc1[n], h1[n], y1);
            h2[n] = fmaf(a2[n], h2[n], Bb2[n] * c2);
            y2    = fmaf(Cc2[n], h2[n], y2);
        }
        const float s   = (y1 + c1) + (y2 + c2);   // SSM outputs + residuals
        const float val = G[idx] * s;              // SiLU gate (precomputed)
        OUT[idx] = add ? (OUT[idx] + val) : val;
    }
}

// ---------------------------------------------------------------------------
// Row LayerNorm (two-pass, matches reference) + residual (xb + xd).
// ---------------------------------------------------------------------------
__global__ __launch_bounds__(256)
void ln_residual(const float* __restrict__ F, const float* __restrict__ xb,
                 const float* __restrict__ xd, const float* __restrict__ gma,
                 const float* __restrict__ bta, float* __restrict__ O)
{
    __shared__ float red[256];
    const int row = blockIdx.x;
    const int t   = threadIdx.x;
    const size_t idx = (size_t)row * D_MODEL + t;
    const float x = F[idx];

    red[t] = x;
    __syncthreads();
    #pragma unroll
    for (int off = 128; off > 0; off >>= 1) {
        if (t < off) red[t] += red[t + off];
        __syncthreads();
    }
    const float mu = red[0] * (1.f / D_MODEL);
    __syncthreads();

    const float dx = x - mu;
    red[t] = dx * dx;
    __syncthreads();
    #pragma unroll
    for (int off = 128; off > 0; off >>= 1) {
        if (t < off) red[t] += red[t + off];
        __syncthreads();
    }
    const float var = red[0] * (1.f / D_MODEL);

    O[idx] = dx * rsqrtf(var + LN_EPS) * gma[t] + bta[t] + xb[idx] + xd[idx];
}

// ---------------------------------------------------------------------------
// out[b,d] = mean over t of O[b,t,d]; coalesced column sum, f64 accumulator.
// ---------------------------------------------------------------------------
__global__ __launch_bounds__(256)
void mean_seq(const float* __restrict__ O, float* __restrict__ out)
{
    const int b = blockIdx.x;
    const int d = threadIdx.x;
    const float* p = O + (size_t)b * SEQ_LEN * D_MODEL + d;
    double s = 0.0;
    for (int t = 0; t < SEQ_LEN; ++t) s += (double)p[(size_t)t * D_MODEL];
    out[b * D_MODEL + d] = (float)(s * (1.0 / SEQ_LEN));
}

// ---------------------------------------------------------------------------
// Input layout (flattened setup_inputs order):
//  0:x_breast 1:x_density
//  per branch (b at 2, d at 22): gate_w, gate_b, mlp_w1, mlp_b1, mlp_w2, mlp_b2,
//    val_w, val_b, conv1_w, conv1_b, ssm1_A, ssm1_delta, ssm1_B, ssm1_C,
//    conv2_w, conv2_b, ssm2_A, ssm2_delta, ssm2_B, ssm2_C
//  42:final_w 43:final_b 44:ln_g 45:ln_b
// ---------------------------------------------------------------------------
extern "C" void kernel_launch(void* const* d_in, const int* in_sizes, int n_in,
                              void* d_out, int out_size, void* d_ws, size_t ws_size,
                              hipStream_t stream)
{
    (void)in_sizes; (void)n_in; (void)out_size; (void)ws_size;
    const float* xb      = (const float*)d_in[0];
    const float* xd      = (const float*)d_in[1];
    const float* final_w = (const float*)d_in[42];
    const float* final_b = (const float*)d_in[43];
    const float* ln_g    = (const float*)d_in[44];
    const float* ln_b    = (const float*)d_in[45];

    float* T0   = (float*)d_ws;                  // MROWS x 1024 (MLP hidden)
    float* T1   = T0 + (size_t)MROWS * 1024;     // MROWS x 256
    float* T2   = T1 + (size_t)MROWS * 256;      // MROWS x 256
    float* COMB = T2 + (size_t)MROWS * 256;      // MROWS x 256

    for (int br = 0; br < 2; ++br) {
        const float* X  = br ? xd : xb;
        void* const* P  = d_in + (br ? 22 : 2);
        // gate path: t = X@gate_w+b ; h = gelu(t@w1+b1) ; g = silu(h@w2+b2)
        launch_gemm<ACT_NONE>(X,  (const float*)P[0], (const float*)P[1], T1, 256, 256,  stream);
        launch_gemm<ACT_GELU>(T1, (const float*)P[2], (const float*)P[3], T0, 1024, 256, stream);
        launch_gemm<ACT_SILU>(T0, (const float*)P[4], (const float*)P[5], T1, 256, 1024, stream);
        // value path
        launch_gemm<ACT_NONE>(X,  (const float*)P[6], (const float*)P[7], T2, 256, 256,  stream);
        // conv + dual SSM + gate, accumulate into COMB across branches
        conv_ssm_gate<<<dim3(BATCH), dim3(D_MODEL), 0, stream>>>(
            T2, T1,
            (const float*)P[8],  (const float*)P[9],  (const float*)P[10], (const float*)P[11],
            (const float*)P[12], (const float*)P[13],
            (const float*)P[14], (const float*)P[15], (const float*)P[16], (const float*)P[17],
            (const float*)P[18], (const float*)P[19],
            COMB, br);
    }

    launch_gemm<ACT_NONE>(COMB, final_w, final_b, T1, 256, 256, stream);
    ln_residual<<<dim3(MROWS), dim3(D_MODEL), 0, stream>>>(T1, xb, xd, ln_g, ln_b, T2);
    mean_seq<<<dim3(BATCH), dim3(D_MODEL), 0, stream>>>(T2, (float*)d_out);
}